// Relut_16595753632503
// MI455X (gfx1250) — compile-verified
//
#include <hip/hip_runtime.h>
#include <hip/hip_bf16.h>

// expm(relu(logm(P))) for batched 64x64 SPD matrices, all-matmul formulation
// mapped to V_WMMA_F32_16X16X4_F32 on gfx1250 (wave32).
//
// v4 changes vs v3:
//  - B-fragment base pointer for the second tile column is laundered into its
//    own VGPR so the compiler's ds_load_2addr_b32 merging pairs (b.x,b.y)
//    (K-row pair, offsets {0,68}) instead of (b0.x,b1.x) (column pair,
//    offsets {0,16}). Loaded pairs land in consecutive VGPRs directly
//    consumable by v_wmma -> removes the v_dual_mov repacking storm and
//    cuts register pressure; DS instruction count unchanged.

typedef float v2f __attribute__((ext_vector_type(2)));
typedef float v4f __attribute__((ext_vector_type(4)));
typedef float v8f __attribute__((ext_vector_type(8)));

#define NDIM   64
#define LDS_LD 68                 // row stride in floats (bank-conflict padding)
#define MATF   (NDIM * LDS_LD)    // floats per LDS matrix buffer
#define NTHR   128                // 4 wave32 per matrix

// Materialize a constant in a VGPR the compiler cannot constant-fold,
// so pointer arithmetic with it yields a distinct base register.
__device__ __forceinline__ int opaque_int(int x) {
  int v;
  asm volatile("v_mov_b32 %0, %1" : "=v"(v) : "i"(x));
  return v;
}

// ---- matmul epilogues (applied per C element at (r,c)) --------------------
struct EpNone {
  __device__ __forceinline__ float operator()(float acc, int, int) const { return acc; }
};
struct EpNS {              // 1.5*Old - 0.5*acc   (Newton-Schulz update)
  const float* Old;
  __device__ __forceinline__ float operator()(float acc, int r, int c) const {
    return 1.5f * Old[r * LDS_LD + c] - 0.5f * acc;
  }
};
struct EpHorner {          // acc*scale + diag*I  (Taylor/Horner step)
  float scale, diag;
  __device__ __forceinline__ float operator()(float acc, int r, int c) const {
    return acc * scale + ((r == c) ? diag : 0.f);
  }
};
struct EpLogRelu {         // relu(32*acc + logs*I)  (final log step + ReLU)
  float logs;
  __device__ __forceinline__ float operator()(float acc, int r, int c) const {
    float v = 32.f * acc + ((r == c) ? logs : 0.f);
    return v > 0.f ? v : 0.f;
  }
};

// ---------------------------------------------------------------------------
// 64x64 fp32 matmul C = ep(A*B), operands in LDS (row stride LDS_LD).
// 4 waves; wave w owns the 32x32 block at (tm0,tn0) = 32*(w>>1), 32*(w&1),
// i.e. a 2x2 grid of 16x16 tiles: A fragments shared across the 2 columns,
// B fragments shared across the 2 rows -> 4 WMMA per {2x b64 A + 2x 2addr B}.
// Fragment layouts per CDNA5 ISA 7.12.2:
//   A 16x4 f32: lane l<16 -> (M=l, K=2*half+{0,1}) in 2 VGPRs (float2)
//   B 4x16 f32: lane l<16 -> (K=2*half+{0,1}, N=l)
//   C/D 16x16 f32: VGPR r -> row r (lanes 0-15) / row r+8 (lanes 16-31)
// ---------------------------------------------------------------------------
template <typename Ep>
__device__ __forceinline__ void mm64(const float* A, const float* B, float* C, Ep ep) {
  const int tid  = threadIdx.x;
  const int wave = tid >> 5;              // 0..3
  const int lane = tid & 31;
  const int half = lane >> 4;
  const int l    = lane & 15;

  const int tm0 = (wave >> 1) << 5;       // 0 or 32
  const int tn0 = (wave & 1) << 5;        // 0 or 32

  v8f acc00 = {0.f,0.f,0.f,0.f,0.f,0.f,0.f,0.f};
  v8f acc01 = {0.f,0.f,0.f,0.f,0.f,0.f,0.f,0.f};
  v8f acc10 = {0.f,0.f,0.f,0.f,0.f,0.f,0.f,0.f};
  v8f acc11 = {0.f,0.f,0.f,0.f,0.f,0.f,0.f,0.f};

  const float* ap0 = A + (tm0 + l) * LDS_LD + 2 * half;
  const float* ap1 = A + (tm0 + 16 + l) * LDS_LD + 2 * half;
  const float* bp0 = B + (2 * half) * LDS_LD + tn0 + l;
  const float* bp1 = bp0 + opaque_int(16);   // own base VGPR (see header note)

#pragma unroll
  for (int kc = 0; kc < 16; ++kc) {        // K = 64 in chunks of 4
    v2f b0, b1;
    b0.x = bp0[kc * 4 * LDS_LD];           // row 4*kc + 2*half,     col tn0+l
    b0.y = bp0[kc * 4 * LDS_LD + LDS_LD];  // row 4*kc + 2*half + 1
    b1.x = bp1[kc * 4 * LDS_LD];           //                        col tn0+16+l
    b1.y = bp1[kc * 4 * LDS_LD + LDS_LD];
    v2f a0 = *(const v2f*)(ap0 + kc * 4);  // rows tm0+l,    K = 4*kc+2*half+{0,1}
    v2f a1 = *(const v2f*)(ap1 + kc * 4);  // rows tm0+16+l
    acc00 = __builtin_amdgcn_wmma_f32_16x16x4_f32(false, a0, false, b0,
                                                  (short)0, acc00, false, false);
    acc01 = __builtin_amdgcn_wmma_f32_16x16x4_f32(false, a0, false, b1,
                                                  (short)0, acc01, false, false);
    acc10 = __builtin_amdgcn_wmma_f32_16x16x4_f32(false, a1, false, b0,
                                                  (short)0, acc10, false, false);
    acc11 = __builtin_amdgcn_wmma_f32_16x16x4_f32(false, a1, false, b1,
                                                  (short)0, acc11, false, false);
  }

  const int row0 = tm0 + 8 * half;
  const int c0   = tn0 + l;
  float* cp0 = C + row0 * LDS_LD;
  float* cp1 = C + (row0 + 16) * LDS_LD;
#pragma unroll
  for (int r = 0; r < 8; ++r) {
    cp0[r * LDS_LD + c0]      = ep(acc00[r], row0 + r,      c0);
    cp0[r * LDS_LD + c0 + 16] = ep(acc01[r], row0 + r,      c0 + 16);
    cp1[r * LDS_LD + c0]      = ep(acc10[r], row0 + 16 + r, c0);
    cp1[r * LDS_LD + c0 + 16] = ep(acc11[r], row0 + 16 + r, c0 + 16);
  }
}

__device__ __forceinline__ float block_reduce_sum(float v, float* red) {
  const int tid = threadIdx.x;
  red[tid] = v;
  __syncthreads();
#pragma unroll
  for (int s = NTHR / 2; s > 0; s >>= 1) {
    if (tid < s) red[tid] += red[tid + s];
    __syncthreads();
  }
  float r = red[0];
  __syncthreads();
  return r;
}

__global__ __launch_bounds__(NTHR) void spd_logrelu_exp_kernel(
    const float* __restrict__ Pg, float* __restrict__ Og) {
  __shared__ float smem[4 * MATF + NTHR];
  float* Y   = smem + 0 * MATF;
  float* Z   = smem + 1 * MATF;
  float* T   = smem + 2 * MATF;
  float* W   = smem + 3 * MATF;
  float* red = smem + 4 * MATF;

  const int tid = threadIdx.x;
  const size_t base = (size_t)blockIdx.x * (NDIM * NDIM);
  const float* Pm = Pg + base;

  // ---- load P -> Y (64x64 row-major -> 64 x LDS_LD) -----------------------
  {
    const int r  = tid >> 1;              // 0..63
    const int c0 = (tid & 1) * 32;        // 0 or 32
#pragma unroll
    for (int k = 0; k < 8; ++k)
      *(v4f*)(Y + r * LDS_LD + c0 + 4 * k) = *(const v4f*)(Pm + r * NDIM + c0 + 4 * k);
  }
  __syncthreads();

  // ---- s = ||P||_F  (>= lambda_max); scale so eigenvalues in (0,1] --------
  float p = 0.f;
  for (int i = tid; i < NDIM * NDIM; i += NTHR) {
    const float v = Y[(i >> 6) * LDS_LD + (i & 63)];
    p += v * v;
  }
  const float s     = sqrtf(block_reduce_sum(p, red));
  const float inv_s = 1.f / s;
  const float logs  = logf(s);
  for (int i = tid; i < NDIM * NDIM; i += NTHR)
    Y[(i >> 6) * LDS_LD + (i & 63)] *= inv_s;
  __syncthreads();

  // ---- 5 repeated sqrts: Y <- Y^(1/2) via coupled Newton-Schulz -----------
  // T = Z*Y;  newY = 1.5Y - 0.5*(Y*T);  newZ = 1.5Z - 0.5*(T*Z)
  const int ns_iters[5] = {15, 12, 10, 9, 8};
  for (int sq = 0; sq < 5; ++sq) {
    for (int i = tid; i < NDIM * NDIM; i += NTHR) {
      const int r = i >> 6, c = i & 63;
      Z[r * LDS_LD + c] = (r == c) ? 1.f : 0.f;
    }
    __syncthreads();
    for (int it = 0; it < ns_iters[sq]; ++it) {
      mm64(Z, Y, T, EpNone{});  __syncthreads();   // T = Z*Y
      mm64(Y, T, W, EpNS{Y});   __syncthreads();   // W = newY (reads old Y)
      mm64(T, Z, Y, EpNS{Z});   __syncthreads();   // oldY buf = newZ (reads Z)
      float* oY = Y; float* oZ = Z;
      Y = W;        // newY
      Z = oY;       // newZ
      W = oZ;       // free
    }
  }
  // Y = M = (P/s)^(1/32), eigenvalues in ~[0.7, 1]

  // ---- log(M) via 8-term Taylor in E = M - I (Horner, fused diag adds) ----
  for (int i = tid; i < NDIM * NDIM; i += NTHR) {
    const int r = i >> 6, c = i & 63;
    T[r * LDS_LD + c] = Y[r * LDS_LD + c] - ((r == c) ? 1.f : 0.f);  // E
  }
  __syncthreads();
  const float ck[9] = {0.f, 1.f, -1.f / 2.f, 1.f / 3.f, -1.f / 4.f,
                       1.f / 5.f, -1.f / 6.f, 1.f / 7.f, -1.f / 8.f};
  float* acc = Y;   // Y contents no longer needed (E saved in T)
  float* tmp = W;
  for (int i = tid; i < NDIM * NDIM; i += NTHR) {
    const int r = i >> 6, c = i & 63;
    acc[r * LDS_LD + c] = (r == c) ? ck[8] : 0.f;
  }
  __syncthreads();
  for (int k = 7; k >= 1; --k) {
    mm64(T, acc, tmp, EpHorner{1.f, ck[k]});  __syncthreads();  // E*acc + ck*I
    float* sw = acc; acc = tmp; tmp = sw;
  }
  // L = relu(32*(E*acc) + log(s)*I), fused into final matmul epilogue
  mm64(T, acc, tmp, EpLogRelu{logs});  __syncthreads();
  float* L = tmp;

  // ---- expm(L): scaling-and-squaring + 8-term Taylor ----------------------
  p = 0.f;
  for (int i = tid; i < NDIM * NDIM; i += NTHR) {
    const float v = L[(i >> 6) * LDS_LD + (i & 63)];
    p += v * v;
  }
  const float fl = sqrtf(block_reduce_sum(p, red));
  int j = 0;
  float sc = 1.f;
  while (j < 12 && fl * sc > 0.5f) { sc *= 0.5f; ++j; }   // uniform across block
  for (int i = tid; i < NDIM * NDIM; i += NTHR)
    L[(i >> 6) * LDS_LD + (i & 63)] *= sc;                // X = L / 2^j
  __syncthreads();

  float* Rk = T;  // E no longer needed
  float* Rt = Z;
  for (int i = tid; i < NDIM * NDIM; i += NTHR) {         // Rk = I + X/8
    const int r = i >> 6, c = i & 63, o = r * LDS_LD + c;
    Rk[o] = 0.125f * L[o] + ((r == c) ? 1.f : 0.f);
  }
  __syncthreads();
  for (int k = 7; k >= 1; --k) {                          // Rt = (X*Rk)/k + I
    mm64(L, Rk, Rt, EpHorner{1.f / (float)k, 1.f});  __syncthreads();
    float* sw = Rk; Rk = Rt; Rt = sw;
  }
  for (int q = 0; q < j; ++q) {                           // undo scaling
    mm64(Rk, Rk, Rt, EpNone{});  __syncthreads();
    float* sw = Rk; Rk = Rt; Rt = sw;
  }

  // ---- store result -------------------------------------------------------
  {
    const int r  = tid >> 1;
    const int c0 = (tid & 1) * 32;
#pragma unroll
    for (int k = 0; k < 8; ++k)
      *(v4f*)(Og + base + r * NDIM + c0 + 4 * k) =
          *(const v4f*)(Rk + r * LDS_LD + c0 + 4 * k);
  }
}

extern "C" void kernel_launch(void* const* d_in, const int* in_sizes, int n_in,
                              void* d_out, int out_size, void* d_ws, size_t ws_size,
                              hipStream_t stream) {
  const float* P = (const float*)d_in[0];
  float* out = (float*)d_out;
  const int nmat = in_sizes[0] / (NDIM * NDIM);   // 1024*16 = 16384 matrices
  spd_logrelu_exp_kernel<<<dim3(nmat), dim3(NTHR), 0, stream>>>(P, out);
}